// SAGERerankerWithNorm_48885317763286
// MI455X (gfx1250) — compile-verified
//
#include <hip/hip_runtime.h>

// Problem constants (match reference)
#define NNODES 50000
#define NEDGES 800000
#define DIN    256
#define DH     128
#define DHALF  64

typedef __attribute__((ext_vector_type(2))) float v2f;
typedef __attribute__((ext_vector_type(8))) float v8f;

// ---------------------------------------------------------------------------
// Utility kernels
// ---------------------------------------------------------------------------
__global__ void zero_kernel(float* __restrict__ p, int total) {
    int i = blockIdx.x * blockDim.x + threadIdx.x;
    if (i < total) p[i] = 0.0f;
}

__global__ void degree_kernel(const int* __restrict__ dst, float* __restrict__ cnt, int e) {
    int i = blockIdx.x * blockDim.x + threadIdx.x;
    if (i < e) atomicAdd(&cnt[dst[i]], 1.0f);
}

// ---------------------------------------------------------------------------
// Multi-B f32 WMMA GEMM:  C_m[M,Nc] = A[M,K] @ B_m[K,Nc]  for m in [0,NMAT)
// One wave handles MT row-tiles x 1 col-tile x NMAT weight matrices, so each
// A fragment (global_load_b64) feeds NMAT v_wmma_f32_16x16x4_f32 ops.
// Requirements: M%16==0, Nc%16==0, K%4==0; EXEC all-ones through the k-loop.
// ---------------------------------------------------------------------------
template<int NMAT, int MT>
__global__ void gemm_wmma_multi(const float* __restrict__ A,
                                const float* __restrict__ B0,
                                const float* __restrict__ B1,
                                const float* __restrict__ B2,
                                const float* __restrict__ bias0,
                                float* __restrict__ C0,
                                float* __restrict__ C1,
                                float* __restrict__ C2,
                                int M, int K, int Nc, int tilesN, int doRelu) {
    const int wave = threadIdx.x >> 5;
    const int lane = threadIdx.x & 31;
    const int tilesM  = M >> 4;
    const int groupsM = (tilesM + MT - 1) / MT;
    const int group   = blockIdx.x * (blockDim.x >> 5) + wave;
    if (group >= groupsM * tilesN) return;     // whole-wave early out

    const int gm = group / tilesN;
    const int tn = group - gm * tilesN;
    const int halfSel = lane >> 4;             // 0: K pair (k,k+1); 1: (k+2,k+3)
    const int l15 = lane & 15;
    const int colB = (tn << 4) + l15;

    const float* __restrict__ Bm[3] = { B0, B1, B2 };
    float*       __restrict__ Cm[3] = { C0, C1, C2 };

    // A-fragment base pointers per row tile (clamped: last group may repeat a tile)
    const float* Arow[MT];
    int tmIdx[MT];
    #pragma unroll
    for (int i = 0; i < MT; ++i) {
        int tm = gm * MT + i;
        tmIdx[i] = tm;
        if (tm > tilesM - 1) tm = tilesM - 1;  // clamp: redundant compute, guarded store
        const int rowA = (tm << 4) + l15;
        Arow[i] = A + (size_t)rowA * K + 2 * halfSel;
    }
    const float* Bcol[NMAT];
    #pragma unroll
    for (int m = 0; m < NMAT; ++m)
        Bcol[m] = Bm[m] + (size_t)(2 * halfSel) * Nc + colB;

    v8f acc[NMAT][MT];
    #pragma unroll
    for (int m = 0; m < NMAT; ++m)
        #pragma unroll
        for (int i = 0; i < MT; ++i)
            acc[m][i] = (v8f){};

    #pragma unroll 2
    for (int k = 0; k < K; k += 4) {
        v2f a[MT];
        #pragma unroll
        for (int i = 0; i < MT; ++i)
            a[i] = *(const v2f*)(Arow[i] + k);       // 8B-aligned global_load_b64
        #pragma unroll
        for (int m = 0; m < NMAT; ++m) {
            v2f b;
            b.x = Bcol[m][(size_t)k * Nc];
            b.y = Bcol[m][(size_t)(k + 1) * Nc];
            #pragma unroll
            for (int i = 0; i < MT; ++i)
                acc[m][i] = __builtin_amdgcn_wmma_f32_16x16x4_f32(
                    false, a[i], false, b, (short)0, acc[m][i], false, false);
        }
    }

    const float bv0 = bias0 ? bias0[colB] : 0.0f;
    #pragma unroll
    for (int i = 0; i < MT; ++i) {
        if (tmIdx[i] >= tilesM) continue;            // uniform per-wave guard
        const int baseRow = (tmIdx[i] << 4) + (halfSel << 3);
        #pragma unroll
        for (int m = 0; m < NMAT; ++m) {
            const float bv = (m == 0) ? bv0 : 0.0f;
            #pragma unroll
            for (int r = 0; r < 8; ++r) {
                float v = acc[m][i][r] + bv;
                if (doRelu) v = fmaxf(v, 0.0f);
                Cm[m][(size_t)(baseRow + r) * Nc + colB] = v;  // coalesced across lanes
            }
        }
    }
}

// ---------------------------------------------------------------------------
// Edge aggregation: AGG[dst] += X[src], rows of DH=128 floats.
// One wave per edge; each lane handles 4 floats -> coalesced 128B bursts.
// ---------------------------------------------------------------------------
__global__ void scatter_add_kernel(const float* __restrict__ X, const int* __restrict__ src,
                                   const int* __restrict__ dst, float* __restrict__ agg, int e) {
    const int edge = (blockIdx.x * blockDim.x + threadIdx.x) >> 5;
    const int lane = threadIdx.x & 31;
    if (edge >= e) return;
    const int s = src[edge];
    const int d = dst[edge];
    const float* __restrict__ xs = X + (size_t)s * DH;
    float* __restrict__ ag = agg + (size_t)d * DH;
    #pragma unroll
    for (int j = 0; j < 4; ++j) {
        atomicAdd(&ag[lane + 32 * j], xs[lane + 32 * j]);
    }
}

// ---------------------------------------------------------------------------
// Fused SAGE combine (float4 wide):
//   out = relu(agg/max(cnt,1) + bias[col] + lin_r) + skip
// ---------------------------------------------------------------------------
__global__ void combine_kernel(const float4* __restrict__ agg, const float* __restrict__ cnt,
                               const float4* __restrict__ bias, const float4* __restrict__ lin_r,
                               const float4* __restrict__ skip, float4* __restrict__ out,
                               int total4) {
    int idx = blockIdx.x * blockDim.x + threadIdx.x;
    if (idx >= total4) return;
    const int node = idx >> 5;            // DH/4 = 32 float4 per row
    const int col4 = idx & 31;
    const float inv = 1.0f / fmaxf(cnt[node], 1.0f);
    const float4 ag = agg[idx];
    const float4 bi = bias[col4];
    const float4 lr = lin_r[idx];
    const float4 sk = skip[idx];
    float4 o;
    o.x = fmaxf(ag.x * inv + bi.x + lr.x, 0.0f) + sk.x;
    o.y = fmaxf(ag.y * inv + bi.y + lr.y, 0.0f) + sk.y;
    o.z = fmaxf(ag.z * inv + bi.z + lr.z, 0.0f) + sk.z;
    o.w = fmaxf(ag.w * inv + bi.w + lr.w, 0.0f) + sk.w;
    out[idx] = o;
}

// ---------------------------------------------------------------------------
// Head tail: out[i] = alpha*rr[i] + (1-alpha)*(dot(T[i,:64], W2) + b2)
// One wave per node, shuffle reduction.
// ---------------------------------------------------------------------------
__global__ void final_kernel(const float* __restrict__ T, const float* __restrict__ W2,
                             const float* __restrict__ b2, const float* __restrict__ rr,
                             const float* __restrict__ alpha_logit, float* __restrict__ out, int n) {
    const int node = (blockIdx.x * blockDim.x + threadIdx.x) >> 5;
    const int lane = threadIdx.x & 31;
    if (node >= n) return;
    const float* __restrict__ t = T + (size_t)node * DHALF;
    float p = t[lane] * W2[lane] + t[lane + 32] * W2[lane + 32];
    #pragma unroll
    for (int off = 16; off > 0; off >>= 1) p += __shfl_down(p, off, 32);
    if (lane == 0) {
        float s = p + b2[0];
        float a = 1.0f / (1.0f + __expf(-alpha_logit[0]));
        out[node] = a * rr[node] + (1.0f - a) * s;
    }
}

// ---------------------------------------------------------------------------
// Launch
// ---------------------------------------------------------------------------
extern "C" void kernel_launch(void* const* d_in, const int* in_sizes, int n_in,
                              void* d_out, int out_size, void* d_ws, size_t ws_size,
                              hipStream_t stream) {
    const float* x    = (const float*)d_in[0];
    const int*   ei   = (const int*)  d_in[1];
    const float* rr   = (const float*)d_in[2];
    const float* Wp   = (const float*)d_in[3];
    const float* bp   = (const float*)d_in[4];
    const float* Wl0  = (const float*)d_in[5];
    const float* bl0  = (const float*)d_in[6];
    const float* Wr0  = (const float*)d_in[7];
    const float* Wl1  = (const float*)d_in[8];
    const float* bl1  = (const float*)d_in[9];
    const float* Wr1  = (const float*)d_in[10];
    const float* W1   = (const float*)d_in[11];
    const float* b1   = (const float*)d_in[12];
    const float* W2   = (const float*)d_in[13];
    const float* b2   = (const float*)d_in[14];
    const float* alog = (const float*)d_in[15];
    float* out = (float*)d_out;

    const int* src = ei;           // edge_index[0]
    const int* dst = ei + NEDGES;  // edge_index[1]

    // Workspace layout (floats): 5 buffers of N*DH + cnt of N  (~128 MB)
    const size_t NB = (size_t)NNODES * DH;
    float* bufR   = (float*)d_ws;       // residual, later h2
    float* bufL   = bufR + NB;          // x@Wl0, later h@Wl1, later head hidden T
    float* bufRt  = bufL + NB;          // x@Wr0, later h@Wr1
    float* bufAgg = bufRt + NB;         // aggregation accumulator (both convs)
    float* bufH   = bufAgg + NB;        // h after conv0
    float* cnt    = bufH + NB;          // degree counts [N]

    const int totalND = NNODES * DH;
    const int total4  = totalND / 4;
    const int ZBLK = 256;

    // degrees + zero agg/cnt
    zero_kernel<<<(NNODES + ZBLK - 1) / ZBLK, ZBLK, 0, stream>>>(cnt, NNODES);
    zero_kernel<<<(totalND + ZBLK - 1) / ZBLK, ZBLK, 0, stream>>>(bufAgg, totalND);
    degree_kernel<<<(NEDGES + ZBLK - 1) / ZBLK, ZBLK, 0, stream>>>(dst, cnt, NEDGES);

    const int gB = 256, wavesPerBlk = gB / 32;
    const int tilesM  = NNODES / 16;            // 3125
    const int groupsM = (tilesM + 1) / 2;       // MT = 2 -> 1563

    // ---- Layer 0: residual/xl0/xr0 share A=x (K=256, Nc=128), x read once ----
    {
        const int nGroups = groupsM * (DH / 16);
        dim3 grid((nGroups + wavesPerBlk - 1) / wavesPerBlk);
        gemm_wmma_multi<3, 2><<<grid, gB, 0, stream>>>(
            x, Wp, Wl0, Wr0, bp, bufR, bufL, bufRt,
            NNODES, DIN, DH, DH / 16, 0);
    }

    // ---- Conv0 aggregate (projected space: mean commutes with linear) ----
    scatter_add_kernel<<<(NEDGES * 32 + gB - 1) / gB, gB, 0, stream>>>(bufL, src, dst, bufAgg, NEDGES);
    combine_kernel<<<(total4 + ZBLK - 1) / ZBLK, ZBLK, 0, stream>>>(
        (const float4*)bufAgg, cnt, (const float4*)bl0, (const float4*)bufRt,
        (const float4*)bufR, (float4*)bufH, total4);

    // ---- Layer 1: hl1/hr1 share A=h (K=128, Nc=128) ----
    zero_kernel<<<(totalND + ZBLK - 1) / ZBLK, ZBLK, 0, stream>>>(bufAgg, totalND);
    {
        const int nGroups = groupsM * (DH / 16);
        dim3 grid((nGroups + wavesPerBlk - 1) / wavesPerBlk);
        gemm_wmma_multi<2, 2><<<grid, gB, 0, stream>>>(
            bufH, Wl1, Wr1, nullptr, nullptr, bufL, bufRt, nullptr,
            NNODES, DH, DH, DH / 16, 0);
    }

    // ---- Conv1 aggregate + combine (result into bufR = h2) ----
    scatter_add_kernel<<<(NEDGES * 32 + gB - 1) / gB, gB, 0, stream>>>(bufL, src, dst, bufAgg, NEDGES);
    combine_kernel<<<(total4 + ZBLK - 1) / ZBLK, ZBLK, 0, stream>>>(
        (const float4*)bufAgg, cnt, (const float4*)bl1, (const float4*)bufRt,
        (const float4*)bufH, (float4*)bufR, total4);

    // ---- Score head: T = relu(h2 @ W1 + b1)  (K=128, Nc=64), fused relu ----
    {
        const int nGroups = groupsM * (DHALF / 16);
        dim3 grid((nGroups + wavesPerBlk - 1) / wavesPerBlk);
        gemm_wmma_multi<1, 2><<<grid, gB, 0, stream>>>(
            bufR, W1, nullptr, nullptr, b1, bufL, nullptr, nullptr,
            NNODES, DH, DHALF, DHALF / 16, 1);
    }

    // ---- Final blend ----
    final_kernel<<<(NNODES * 32 + gB - 1) / gB, gB, 0, stream>>>(bufL, W2, b2, rr, alog, out, NNODES);
}